// BottomUpIntegrator_39900246180024
// MI455X (gfx1250) — compile-verified
//
#include <hip/hip_runtime.h>

// BottomUpIntegrator for MI455X (gfx1250, wave32).
// Pass 1: per-cell MLP via V_WMMA_F32_16X16X4_F32 (fp32 matrix path) + sorted-segment
//         wave-combined atomics into cluster accumulators.
// Pass 2: per-cluster stats + tiny cluster MLP.
// Pass 3: global archetype reduction.

#define NCLUSTERS 8192
#define SDIM 32
#define ADIM 4
#define HDIM 64
#define XDIM 36  // SDIM + ADIM

typedef float v2f __attribute__((ext_vector_type(2)));
typedef float v8f __attribute__((ext_vector_type(8)));

__global__ __launch_bounds__(256) void zero_ws_kernel(float* __restrict__ ws, int n) {
  int i = blockIdx.x * 256 + threadIdx.x;
  if (i < n) ws[i] = 0.0f;
}

__global__ __launch_bounds__(256) void cell_pass_kernel(
    const float* __restrict__ cell_state,
    const float* __restrict__ arch_state,
    const float* __restrict__ energy,
    const float* __restrict__ phi_local,
    const float* __restrict__ W1,
    const float* __restrict__ b1,
    const float* __restrict__ W2,
    const float* __restrict__ b2,
    const int*   __restrict__ seg_ids,
    float* __restrict__ acc_counts,
    float* __restrict__ acc_wsum,
    float* __restrict__ acc_warch,
    float* __restrict__ acc_arch,
    float* __restrict__ acc_arch2,
    int n_cells)
{
  __shared__ float sW1[XDIM * HDIM];  // [36][64] row-major
  __shared__ float sB1[HDIM];
  __shared__ float sW2[HDIM];
  __shared__ float sB2[1];

  for (int i = threadIdx.x; i < XDIM * HDIM; i += 256) sW1[i] = W1[i];
  if (threadIdx.x < HDIM) {
    sB1[threadIdx.x] = b1[threadIdx.x];
    sW2[threadIdx.x] = W2[threadIdx.x];
  }
  if (threadIdx.x == 0) sB2[0] = b2[0];
  __syncthreads();

  const int lane   = threadIdx.x & 31;
  const int col    = lane & 15;   // N column (and A-row M) this lane serves
  const int hiHalf = lane >> 4;   // 0: K+{0,1}, 1: K+{2,3} per ISA 16x4 f32 layout
  const int waveId = (blockIdx.x * blockDim.x + threadIdx.x) >> 5;
  const int nWaves = (gridDim.x * blockDim.x) >> 5;

  // per-lane column constants (N = nt*16 + col), hoisted out of the tile loop
  float b1c[4], w2c[4];
#pragma unroll
  for (int nt = 0; nt < 4; ++nt) {
    b1c[nt] = sB1[nt * 16 + col];
    w2c[nt] = sW2[nt * 16 + col];
  }
  const float bias2 = sB2[0];

  const int nTiles = n_cells >> 4;  // 16 cells per wave-tile
  for (int tile = waveId; tile < nTiles; tile += nWaves) {
    const int base  = tile << 4;
    const int cellA = base + col;  // row M fed by this lane into the A fragment

    const v8f vzero = {0.f, 0.f, 0.f, 0.f, 0.f, 0.f, 0.f, 0.f};
    v8f accs[4] = {vzero, vzero, vzero, vzero};

    // h[16x64] = x[16x36] @ W1[36x64] : 9 K-steps x 4 N-tiles of 16x16x4 f32 WMMA
#pragma unroll
    for (int kk = 0; kk < 9; ++kk) {
      const int k0 = kk * 4 + hiHalf * 2;  // K indices {k0, k0+1} for this lane
      v2f afrag;
      if (kk < 8) {  // compile-time branch (fully unrolled)
        const float* p = cell_state + (size_t)cellA * SDIM + k0;
        afrag[0] = p[0];
        afrag[1] = p[1];
      } else {       // last K-step covers the 4 arch_state features
        const float* p = arch_state + (size_t)cellA * ADIM + (k0 - SDIM);
        afrag[0] = p[0];
        afrag[1] = p[1];
      }
#pragma unroll
      for (int nt = 0; nt < 4; ++nt) {
        v2f bfrag;  // B (4x16): same K pair, column N = nt*16+col
        bfrag[0] = sW1[(k0 + 0) * HDIM + nt * 16 + col];
        bfrag[1] = sW1[(k0 + 1) * HDIM + nt * 16 + col];
        accs[nt] = __builtin_amdgcn_wmma_f32_16x16x4_f32(
            false, afrag, false, bfrag, (short)0, accs[nt], false, false);
      }
    }

    // Layer 2: y[M] = relu(h[M]+b1) . W2 + b2.
    // C/D layout: VGPR m holds row M = m + 8*hiHalf, this lane's column N = nt*16+col.
    float part[8];
#pragma unroll
    for (int m = 0; m < 8; ++m) {
      float s = 0.0f;
#pragma unroll
      for (int nt = 0; nt < 4; ++nt) {
        float h = accs[nt][m] + b1c[nt];
        h = fmaxf(h, 0.0f);
        s += h * w2c[nt];
      }
      part[m] = s;
    }
    // reduce the 64 columns: butterfly over the 16 lanes of each half-wave
#pragma unroll
    for (int off = 1; off < 16; off <<= 1) {
#pragma unroll
      for (int m = 0; m < 8; ++m) part[m] += __shfl_xor(part[m], off, 32);
    }

    // Cell ownership: lanes 0-7 -> cells 0-7, lanes 16-23 -> cells 8-15.
    // Lanes 8-15/24-31 mirror their partner (same reduced sums) so every lane
    // has well-defined values for the wave-uniformity test below.
    const int  myM     = hiHalf * 8 + (lane & 7);
    const int  myCell  = base + myM;
    const bool handler = ((lane & 8) == 0);

    float y = part[0];
    const int sel = lane & 7;
#pragma unroll
    for (int m = 1; m < 8; ++m) y = (sel == m) ? part[m] : y;
    y += bias2;

    const float gate = 1.0f / (1.0f + __expf(-y));
    const float e    = energy[myCell];
    const float p    = phi_local[myCell];
    const float imp  = fminf(fmaxf(gate * e * p, 0.01f), 1.0f);
    const float w    = imp * e * p;

    float ar[4];
#pragma unroll
    for (int a = 0; a < 4; ++a) ar[a] = arch_state[(size_t)myCell * ADIM + a];

    const int seg  = seg_ids[myCell];
    const int seg0 = __shfl(seg, 0, 32);
    if (__all(seg == seg0)) {
      // segment_ids are sorted -> common case: whole tile in one cluster.
      // Combine in-wave, then 14 atomics instead of 16x14.
      float vals[14];
      const float mk = handler ? 1.0f : 0.0f;
      vals[0] = mk;
      vals[1] = mk * w;
#pragma unroll
      for (int a = 0; a < 4; ++a) {
        vals[2 + a]  = mk * w * ar[a];
        vals[6 + a]  = mk * ar[a];
        vals[10 + a] = mk * ar[a] * ar[a];
      }
#pragma unroll
      for (int off = 1; off < 32; off <<= 1) {
#pragma unroll
        for (int i = 0; i < 14; ++i) vals[i] += __shfl_xor(vals[i], off, 32);
      }
      if (lane == 0) {
        atomicAdd(acc_counts + seg0, vals[0]);
        atomicAdd(acc_wsum   + seg0, vals[1]);
#pragma unroll
        for (int a = 0; a < 4; ++a) {
          atomicAdd(acc_warch + seg0 * 4 + a, vals[2 + a]);
          atomicAdd(acc_arch  + seg0 * 4 + a, vals[6 + a]);
          atomicAdd(acc_arch2 + seg0 * 4 + a, vals[10 + a]);
        }
      }
    } else if (handler) {  // cluster-boundary tile: per-cell atomics
      atomicAdd(acc_counts + seg, 1.0f);
      atomicAdd(acc_wsum   + seg, w);
#pragma unroll
      for (int a = 0; a < 4; ++a) {
        atomicAdd(acc_warch + seg * 4 + a, w * ar[a]);
        atomicAdd(acc_arch  + seg * 4 + a, ar[a]);
        atomicAdd(acc_arch2 + seg * 4 + a, ar[a] * ar[a]);
      }
    }
  }
}

__global__ __launch_bounds__(256) void cluster_pass_kernel(
    const float* __restrict__ acc_counts,
    const float* __restrict__ acc_wsum,
    const float* __restrict__ acc_warch,
    const float* __restrict__ acc_arch,
    const float* __restrict__ acc_arch2,
    const float* __restrict__ Wc1,
    const float* __restrict__ bc1,
    const float* __restrict__ Wc2,
    const float* __restrict__ bc2,
    float* __restrict__ out_aggregate,
    float* __restrict__ out_phi,
    float* __restrict__ out_coh,
    float* __restrict__ wc_arr,
    float* __restrict__ scalars,  // [0]=wc_sum [1]=nvalid
    int n_clusters)
{
  const int c = blockIdx.x * 256 + threadIdx.x;
  if (c >= n_clusters) return;

  const float cnt  = acc_counts[c];
  const float wsum = acc_wsum[c];
  const float safe = fmaxf(cnt, 1.0f);

  float agg[4];
  float sq = 0.0f;
#pragma unroll
  for (int a = 0; a < 4; ++a) {
    const float sa  = acc_arch[c * 4 + a];
    const float sa2 = acc_arch2[c * 4 + a];
    const float swa = acc_warch[c * 4 + a];
    agg[a] = (wsum > 0.0f) ? swa / fmaxf(wsum, 1e-30f) : sa / safe;
    sq += fmaxf(sa2 - sa * sa / safe, 0.0f);  // sum of squared deviations, dim a
  }
  const float var  = (cnt >= 2.0f) ? (sq * 0.25f) / fmaxf(cnt - 1.0f, 1.0f) : 0.0f;
  const float phic = 1.0f - fminf(1.0f, var * 2.0f);
  const float coh  = 1.0f - var;

  const float feats[7] = {agg[0], agg[1], agg[2], agg[3],
                          phic, coh, fminf(1.0f, cnt * 0.05f)};
  float yc = bc2[0];
  for (int j = 0; j < 32; ++j) {
    float h = bc1[j];
#pragma unroll
    for (int i = 0; i < 7; ++i) h += feats[i] * Wc1[i * 32 + j];
    yc += fmaxf(h, 0.0f) * Wc2[j];
  }
  const float basec = 1.0f / (1.0f + __expf(-yc));
  const float impc  = fminf(fmaxf(basec * phic, 0.01f), 1.0f);
  const bool  valid = cnt > 0.0f;
  const float wcv   = valid ? impc * cnt : 0.0f;

#pragma unroll
  for (int a = 0; a < 4; ++a) out_aggregate[c * 4 + a] = agg[a];
  out_phi[c] = phic;
  out_coh[c] = coh;
  wc_arr[c]  = wcv;
  atomicAdd(&scalars[0], wcv);
  atomicAdd(&scalars[1], valid ? 1.0f : 0.0f);
}

__global__ __launch_bounds__(256) void finalize_kernel(
    const float* __restrict__ wc_arr,
    const float* __restrict__ acc_counts,
    const float* __restrict__ aggregate,
    const float* __restrict__ scalars,
    float* __restrict__ out_global,
    int n_clusters)
{
  __shared__ float red[256 * 4];
  const float wcs = scalars[0];
  const float nv  = fmaxf(scalars[1], 1.0f);

  float g[4] = {0.f, 0.f, 0.f, 0.f};
  for (int c = threadIdx.x; c < n_clusters; c += 256) {
    float nw;
    if (wcs > 0.0f) nw = wc_arr[c] / fmaxf(wcs, 1e-30f);
    else            nw = (acc_counts[c] > 0.0f) ? (1.0f / nv) : 0.0f;
#pragma unroll
    for (int a = 0; a < 4; ++a) g[a] += nw * aggregate[c * 4 + a];
  }
#pragma unroll
  for (int a = 0; a < 4; ++a) red[threadIdx.x * 4 + a] = g[a];
  __syncthreads();
  if (threadIdx.x == 0) {
    float t[4] = {0.f, 0.f, 0.f, 0.f};
    for (int i = 0; i < 256; ++i) {
#pragma unroll
      for (int a = 0; a < 4; ++a) t[a] += red[i * 4 + a];
    }
#pragma unroll
    for (int a = 0; a < 4; ++a) out_global[a] = t[a];
  }
}

extern "C" void kernel_launch(void* const* d_in, const int* in_sizes, int n_in,
                              void* d_out, int out_size, void* d_ws, size_t ws_size,
                              hipStream_t stream) {
  (void)n_in; (void)out_size; (void)ws_size;

  const float* cell_state = (const float*)d_in[0];
  const float* arch_state = (const float*)d_in[1];
  const float* energy     = (const float*)d_in[2];
  const float* phi_local  = (const float*)d_in[3];
  const float* W1  = (const float*)d_in[4];
  const float* b1  = (const float*)d_in[5];
  const float* W2  = (const float*)d_in[6];
  const float* b2  = (const float*)d_in[7];
  const float* Wc1 = (const float*)d_in[8];
  const float* bc1 = (const float*)d_in[9];
  const float* Wc2 = (const float*)d_in[10];
  const float* bc2 = (const float*)d_in[11];
  const int*   seg = (const int*)d_in[12];

  const int N = in_sizes[2];   // energy is [N]
  const int C = NCLUSTERS;     // reference constant (d_in[13] is device mem)

  // Workspace layout (floats): counts | wsum | warch[4C] | arch[4C] | arch2[4C] | wc | scalars[2]
  float* ws         = (float*)d_ws;
  float* acc_counts = ws;
  float* acc_wsum   = ws + C;
  float* acc_warch  = ws + 2 * C;
  float* acc_arch   = ws + 6 * C;
  float* acc_arch2  = ws + 10 * C;
  float* wc_arr     = ws + 14 * C;
  float* scalars    = ws + 15 * C;

  float* out           = (float*)d_out;
  float* out_aggregate = out;            // [C,4]
  float* out_phi       = out + 4 * C;    // [C]
  float* out_coh       = out + 5 * C;    // [C]
  float* out_global    = out + 6 * C;    // [4]

  const int nzero = 15 * C + 2;
  zero_ws_kernel<<<(nzero + 255) / 256, 256, 0, stream>>>(ws, nzero);

  cell_pass_kernel<<<1024, 256, 0, stream>>>(
      cell_state, arch_state, energy, phi_local, W1, b1, W2, b2, seg,
      acc_counts, acc_wsum, acc_warch, acc_arch, acc_arch2, N);

  cluster_pass_kernel<<<(C + 255) / 256, 256, 0, stream>>>(
      acc_counts, acc_wsum, acc_warch, acc_arch, acc_arch2,
      Wc1, bc1, Wc2, bc2, out_aggregate, out_phi, out_coh, wc_arr, scalars, C);

  finalize_kernel<<<1, 256, 0, stream>>>(wc_arr, acc_counts, out_aggregate,
                                         scalars, out_global, C);
}